// MultiAvg_8315056685518
// MI455X (gfx1250) — compile-verified
//
#include <hip/hip_runtime.h>
#include <stdint.h>

// ---- problem constants (match reference module) ----
#define KK        4
#define STRIDE    2
#define PAD       1
#define NSEG      8
#define FRAME_FLOATS 25088      // C*H*W = 128*14*14
#define FRAME_F4     6272       // float4 per frame
#define CHUNK_F4     256        // float4 per pixel chunk (1024 floats = 4 KB/row)
#define CHUNK_FLOATS 1024
#define NCHUNK       25         // ceil(6272/256), last chunk = 128 float4
#define FOUT         8          // output frames per block (divides every segment out-len)
#define MAXROWS      (STRIDE*FOUT + (KK - STRIDE))   // 18 input frames max per block

typedef uint32_t u32x4 __attribute__((ext_vector_type(4)));
typedef uint32_t u32x8 __attribute__((ext_vector_type(8)));
typedef float    f32x4 __attribute__((ext_vector_type(4)));   // native vec for NT stores

__global__ void __launch_bounds__(256)
pool_tdm_kernel(const float* __restrict__ x,
                const int* __restrict__ vid_lens,
                float* __restrict__ out) {
    __shared__ f32x4 tile[MAXROWS * CHUNK_F4];    // 73728 bytes

    const int chunk = blockIdx.x;                 // pixel chunk
    const int n0    = blockIdx.y * FOUT;          // first global output frame

    // ---- locate owning segment (8 segments, scalar loop, keep rolled) ----
    int in_off = 0, L = 0, o0 = 0;
    {
        int io = 0, oo = 0;
        #pragma unroll 1
        for (int i = 0; i < NSEG; ++i) {
            int Li = vid_lens[i];
            int ol = (Li + 2 * PAD - KK) / STRIDE + 1;
            if (n0 >= oo && n0 < oo + ol) { in_off = io; L = Li; o0 = n0 - oo; }
            io += Li; oo += ol;
        }
    }

    const int chunk_f4     = min(CHUNK_F4, FRAME_F4 - chunk * CHUNK_F4);
    const int chunk_floats = chunk_f4 * 4;

    // valid input-frame window covered by this block's 8 outputs (clipped to segment)
    const int jlo   = max(0, STRIDE * o0 - PAD);
    const int jhi   = min(L - 1, STRIDE * o0 - PAD + STRIDE * (FOUT - 1) + KK - 1);
    const int nrows = jhi - jlo + 1;              // <= MAXROWS

    // ---- Tensor Data Mover: 2D tile (nrows frames x chunk_floats f32) -> LDS ----
    const uint64_t gaddr = (uint64_t)(uintptr_t)x +
        (((uint64_t)(in_off + jlo)) * FRAME_FLOATS + (uint64_t)chunk * CHUNK_FLOATS) * 4ull;
    const uint32_t lds_base = (uint32_t)(uintptr_t)&tile[0];   // low 32 bits = LDS byte addr

    // one wave issues the DMA (TDM ignores EXEC -> must be a scalar-branch guard)
    const uint32_t wave_id = __builtin_amdgcn_readfirstlane(threadIdx.x) >> 5;
    if (wave_id == 0) {
        // D# group 0 (128b): count=1 | lds_addr | global_addr[56:0] | type=2
        u32x4 g0;
        g0.x = 1u;
        g0.y = lds_base;
        g0.z = (uint32_t)gaddr;
        g0.w = ((uint32_t)(gaddr >> 32) & 0x01FFFFFFu) | (2u << 30);

        // D# group 1 (256b): no multicast, data_size=4B, no pad/iterate/barrier
        const uint32_t td0 = 0x7FFFFFFFu, td1 = 0x7FFFFFFFu;   // huge dims: no OOB clip
        u32x8 g1;
        g1.s0 = (2u << 16);                                    // data_size code 2 (4B)
        g1.s1 = (td0 & 0xFFFFu) << 16;                         // tensor_dim0[15:0]
        g1.s2 = ((td0 >> 16) & 0xFFFFu) | ((td1 & 0xFFFFu) << 16);
        g1.s3 = ((td1 >> 16) & 0xFFFFu) |
                (((uint32_t)chunk_floats & 0xFFFFu) << 16);    // tile_dim0
        g1.s4 = ((uint32_t)nrows & 0xFFFFu);                   // tile_dim1 (tile_dim2=0)
        g1.s5 = (uint32_t)FRAME_FLOATS;                        // tensor_dim0_stride[31:0]
        g1.s6 = 0u;                                            // stride[47:32] | d1stride lo
        g1.s7 = 0u;

        // 2D tensor -> 2-group form (VADDR2/VADDR3 = NULL); default RT temporal hint
        // so the 2-frame overlap with the neighboring out-chunk stays hot in L2.
        asm volatile("tensor_load_to_lds %0, %1" :: "s"(g0), "s"(g1) : "memory");
        __builtin_amdgcn_s_wait_tensorcnt(0);
    }
    __syncthreads();   // waves 1..7 wait until wave 0 has landed the tile

    // ---- compute: masked average of <=4 taps per output frame ----
    // Output is write-once: store non-temporal (th:TH_STORE_NT) so the 103 MB of
    // results don't evict the reused input lines from the 192 MB L2.
    const int tid = threadIdx.x;
    if (tid < chunk_f4) {
        const size_t col = (size_t)chunk * CHUNK_F4 + tid;
        f32x4* __restrict__ out4 = (f32x4*)out;
        #pragma unroll
        for (int of = 0; of < FOUT; ++of) {
            const int jbase = STRIDE * (o0 + of) - PAD;
            f32x4 acc = (f32x4)(0.f);
            #pragma unroll
            for (int t = 0; t < KK; ++t) {
                const int j = jbase + t;               // block-uniform predicate
                if (j >= 0 && j < L) {
                    acc += tile[(j - jlo) * chunk_f4 + tid];
                }
            }
            acc *= 0.25f;
            __builtin_nontemporal_store(acc, &out4[(size_t)(n0 + of) * FRAME_F4 + col]);
        }
    }
}

// append new_vid_lens (as f32, d_out is a flat f32 buffer) after the pooled tensor
__global__ void lens_kernel(const int* __restrict__ vid_lens, float* __restrict__ out) {
    if (threadIdx.x == 0 && blockIdx.x == 0) {
        long long tot = 0;
        int ol[NSEG];
        for (int i = 0; i < NSEG; ++i) {
            ol[i] = (vid_lens[i] + 2 * PAD - KK) / STRIDE + 1;
            tot += ol[i];
        }
        const size_t base = (size_t)tot * FRAME_FLOATS;
        for (int i = 0; i < NSEG; ++i) out[base + i] = (float)ol[i];
    }
}

extern "C" void kernel_launch(void* const* d_in, const int* in_sizes, int n_in,
                              void* d_out, int out_size, void* d_ws, size_t ws_size,
                              hipStream_t stream) {
    const float* x  = (const float*)d_in[0];
    const int*   vl = (const int*)d_in[1];
    float*       out = (float*)d_out;

    const int n_frames = in_sizes[0] / FRAME_FLOATS;   // 2048
    const int n_out    = n_frames / 2;                 // stride-2, pad-1, k-4 => L/2 per seg
    dim3 grid(NCHUNK, n_out / FOUT);                   // 25 x 128
    pool_tdm_kernel<<<grid, 256, 0, stream>>>(x, vl, out);
    lens_kernel<<<1, 32, 0, stream>>>(vl, out);
}